// MultiVectorQuantizer_46643344834661
// MI455X (gfx1250) — compile-verified
//
#include <hip/hip_runtime.h>
#include <math.h>

typedef __attribute__((ext_vector_type(2))) float v2f;
typedef __attribute__((ext_vector_type(4))) float v4f;
typedef __attribute__((ext_vector_type(8))) float v8f;
typedef __attribute__((ext_vector_type(4))) unsigned int u32x4;
typedef __attribute__((ext_vector_type(8))) int i32x8;
typedef __attribute__((ext_vector_type(4))) int i32x4;

#define T_TOT 8192      // B*L
#define D_TOT 1024      // N*d
#define NHEAD 8
#define KCODE 4096
#define DSUB  128
#define NTILE (KCODE / 16)      // 256 K-tiles
#define LROW  132               // 128 floats + 4-float pad (bank-conflict-free)

// workspace layout (floats):
// [0] commit-sum  [1] entropy-sum  [2 .. 2+N*K) counts  [2+N*K ..) e2 norms
#define WS_COMMIT  0
#define WS_ENTROPY 1
#define WS_COUNTS  2
#define WS_E2      (2 + NHEAD * KCODE)

#if __has_builtin(__builtin_amdgcn_tensor_load_to_lds) && \
    __has_builtin(__builtin_amdgcn_s_wait_tensorcnt)
#define USE_TDM 1
#else
#define USE_TDM 0
#endif

__global__ void mvq_zero_ws(float* __restrict__ ws) {
    int i = blockIdx.x * 256 + threadIdx.x;
    if (i < 2 + NHEAD * KCODE) ws[i] = 0.0f;
}

// ||codebook[n,k,:]||^2 : one wave per (n,k) row, lane loads 4 floats.
__global__ __launch_bounds__(256) void mvq_e2(const float* __restrict__ cb,
                                              float* __restrict__ ws) {
    int wave = threadIdx.x >> 5;
    int lane = threadIdx.x & 31;
    int row  = blockIdx.x * 8 + wave;              // 0 .. N*K-1
    v4f v = *(const v4f*)(cb + (size_t)row * DSUB + lane * 4);
    float s = v.x * v.x + v.y * v.y + v.z * v.z + v.w * v.w;
    #pragma unroll
    for (int m = 16; m >= 1; m >>= 1) s += __shfl_xor(s, m, 32);
    if (lane == 0) ws[WS_E2 + row] = s;
}

#if USE_TDM
// TDM: DMA one 16-code x 128-float tile (8KB) into LDS with a 4-DWORD pad
// inserted every 128 DWORDs (row stride becomes 132 floats in LDS).
__device__ __forceinline__ void tdm_load_tile(const float* gsrc, unsigned lds_off) {
    unsigned long long ga = (unsigned long long)(uintptr_t)gsrc;
    u32x4 g0 = {
        1u,                                               // count=1, user load
        lds_off,                                          // lds_addr (bytes)
        (unsigned)ga,                                     // global_addr[31:0]
        (unsigned)((ga >> 32) & 0x01FFFFFFu) | (2u << 30) // addr[56:32] | type=2
    };
    i32x8 g1 = {
        (2 << 16) | (1 << 20) | (6 << 22) | (3 << 25), // data_size=4B, pad_en,
                                                       // interval=128dw, amount=4dw
        (DSUB << 16),         // tensor_dim0 = 128  (bits 79:48)
        (KCODE << 16),        // tensor_dim1 = 4096 (bits 111:80)
        (DSUB << 16),         // tile_dim0 = 128    (bits 127:112)
        16,                   // tile_dim1 = 16     (bits 143:128)
        DSUB,                 // tensor_dim0_stride = 128 (bits 207:160)
        0, 0
    };
    i32x4 z4 = {0, 0, 0, 0};
    i32x8 z8 = {0, 0, 0, 0, 0, 0, 0, 0};
    __builtin_amdgcn_tensor_load_to_lds(g0, g1, z4, z4, z8, 0);
}
#endif

// Main kernel: workgroup = 8 waves = 128 tokens of one head. Codebook tiles
// (16 codes x 128d) staged in LDS (TDM double-buffered), shared by all waves.
// Per wave: 16-token A tile resident in VGPRs, 32x v_wmma_f32_16x16x4_f32 per
// K-tile into two independent accumulators; running argmin of (e2_k - 2*dot).
__global__ __launch_bounds__(256, 1)
void mvq_main(const float* __restrict__ z, const float* __restrict__ cb,
              float* __restrict__ out, float* __restrict__ ws) {
    __shared__ float lbuf[2][16 * LROW];

    const int n    = blockIdx.y;
    const int wave = threadIdx.x >> 5;
    const int lane = threadIdx.x & 31;
    const int half = lane >> 4;        // 0: M rows 0-7, 1: M rows 8-15
    const int g    = lane & 15;        // N column within tile / A row
    const int tokenBase = blockIdx.x * 128 + wave * 16;

    // ---- Preload A fragments: lane holds z[tokenBase+g][4t + 2*half + {0,1}]
    const float* zrowA = z + (size_t)(tokenBase + g) * D_TOT + n * DSUB + 2 * half;
    v2f areg[32];
    #pragma unroll
    for (int t = 0; t < 32; ++t)
        areg[t] = *(const v2f*)(zrowA + 4 * t);

    const float* cbn = cb + (size_t)n * KCODE * DSUB;
    const float* e2n = ws + WS_E2 + n * KCODE;

    float bestVal[8];
    int   bestIdx[8];
    #pragma unroll
    for (int r = 0; r < 8; ++r) { bestVal[r] = INFINITY; bestIdx[r] = 0; }

    // Per-tile compute from LDS buffer, K-tile index i.
    auto compute_tile = [&](int i) {
        const float* bbase = &lbuf[i & 1][g * LROW + 2 * half];
        v8f acc0 = {0.f, 0.f, 0.f, 0.f, 0.f, 0.f, 0.f, 0.f};
        v8f acc1 = {0.f, 0.f, 0.f, 0.f, 0.f, 0.f, 0.f, 0.f};
        #pragma unroll
        for (int t = 0; t < 32; t += 2) {
            v2f b0 = *(const v2f*)(bbase + 4 * t);
            v2f b1 = *(const v2f*)(bbase + 4 * (t + 1));
            acc0 = __builtin_amdgcn_wmma_f32_16x16x4_f32(
                false, areg[t],     false, b0, (short)0, acc0, false, false);
            acc1 = __builtin_amdgcn_wmma_f32_16x16x4_f32(
                false, areg[t + 1], false, b1, (short)0, acc1, false, false);
        }
        const int   k0   = i * 16;
        const float e2v  = e2n[k0 + g];
        const int   kidx = k0 + g;
        #pragma unroll
        for (int r = 0; r < 8; ++r) {
            float s = fmaf(-2.0f, acc0[r] + acc1[r], e2v);  // ||z||^2 dropped
            if (s < bestVal[r]) { bestVal[r] = s; bestIdx[r] = kidx; }
        }
    };

#if USE_TDM
    const unsigned lds0 = (unsigned)(uintptr_t)&lbuf[0][0];
    const unsigned lds1 = (unsigned)(uintptr_t)&lbuf[1][0];
    if (wave == 0) {
        tdm_load_tile(cbn, lds0);
        __builtin_amdgcn_s_wait_tensorcnt(0);
    }
    __syncthreads();
    for (int i = 0; i < NTILE; ++i) {
        if (wave == 0 && i + 1 < NTILE)
            tdm_load_tile(cbn + (size_t)(i + 1) * 16 * DSUB,
                          ((i + 1) & 1) ? lds1 : lds0);
        compute_tile(i);
        if (wave == 0) __builtin_amdgcn_s_wait_tensorcnt(0); // tile i+1 landed
        __syncthreads();  // release next buffer to everyone / protect reuse
    }
#else
    // Fallback: coalesced reg->LDS double-buffered staging (b128 both sides).
    v4f stg[2];
    auto stage_load = [&](int i) {
        const float* src = cbn + (size_t)i * 16 * DSUB;
        #pragma unroll
        for (int p = 0; p < 2; ++p) {
            int li = threadIdx.x + 256 * p;
            stg[p] = *(const v4f*)(src + (li >> 5) * DSUB + (li & 31) * 4);
        }
    };
    auto stage_store = [&](int b) {
        #pragma unroll
        for (int p = 0; p < 2; ++p) {
            int li = threadIdx.x + 256 * p;
            *(v4f*)&lbuf[b][(li >> 5) * LROW + (li & 31) * 4] = stg[p];
        }
    };
    stage_load(0); stage_store(0); __syncthreads();
    for (int i = 0; i < NTILE; ++i) {
        if (i + 1 < NTILE) stage_load(i + 1);
        compute_tile(i);
        __syncthreads();
        if (i + 1 < NTILE) stage_store((i + 1) & 1);
        __syncthreads();
    }
#endif

    // ---- cross-lane argmin within each 16-lane N group (ties -> lowest index)
    #pragma unroll
    for (int r = 0; r < 8; ++r) {
        float v = bestVal[r]; int idx = bestIdx[r];
        #pragma unroll
        for (int m = 1; m <= 8; m <<= 1) {
            float ov = __shfl_xor(v, m, 32);
            int   oi = __shfl_xor(idx, m, 32);
            if (ov < v || (ov == v && oi < idx)) { v = ov; idx = oi; }
        }
        bestIdx[r] = idx;
    }

    // ---- epilogue: gather zq, write z_q & indices, commit MSE, counts
    float* outZq  = out;
    float* outIdx = out + (size_t)T_TOT * D_TOT;
    float commitLocal = 0.0f;
    #pragma unroll
    for (int r = 0; r < 8; ++r) {
        const int tRow = tokenBase + r + 8 * half;
        const int idx  = bestIdx[r];               // same in all 16 lanes of group
        const float* cq   = cbn + (size_t)idx * DSUB;
        const float* zrow = z     + (size_t)tRow * D_TOT + n * DSUB;
        float*       orow = outZq + (size_t)tRow * D_TOT + n * DSUB;
        #pragma unroll
        for (int s = 0; s < 8; ++s) {
            int j = g + 16 * s;
            float c  = cq[j];
            float zv = zrow[j];
            orow[j] = c;                            // forward z_q == zq
            float dlt = zv - c;
            commitLocal = fmaf(dlt, dlt, commitLocal);
        }
        if (g == 0) {
            outIdx[(size_t)tRow * NHEAD + n] = (float)idx;
            atomicAdd(&ws[WS_COUNTS + n * KCODE + idx], 1.0f);
        }
    }
    #pragma unroll
    for (int m = 16; m >= 1; m >>= 1)
        commitLocal += __shfl_xor(commitLocal, m, 32);
    if (lane == 0) atomicAdd(&ws[WS_COMMIT], commitLocal);
}

// Per-head entropy of code usage; summed into ws[WS_ENTROPY].
__global__ void mvq_entropy(float* __restrict__ ws) {
    __shared__ float red[256];
    const int n = blockIdx.x;
    float local = 0.0f;
    for (int k = threadIdx.x; k < KCODE; k += 256) {
        float p = ws[WS_COUNTS + n * KCODE + k] * (1.0f / (float)T_TOT);
        local -= p * logf(p + 1e-10f);
    }
    red[threadIdx.x] = local;
    __syncthreads();
    for (int s = 128; s > 0; s >>= 1) {
        if (threadIdx.x < (unsigned)s) red[threadIdx.x] += red[threadIdx.x + s];
        __syncthreads();
    }
    if (threadIdx.x == 0) atomicAdd(&ws[WS_ENTROPY], red[0]);
}

__global__ void mvq_finalize(const float* __restrict__ ws, float* __restrict__ out) {
    float commit  = 0.25f * ws[WS_COMMIT] / (float)((size_t)T_TOT * D_TOT);
    float entropy = ws[WS_ENTROPY] / (float)NHEAD;
    size_t base = (size_t)T_TOT * D_TOT + (size_t)T_TOT * NHEAD;
    out[base + 0] = commit;    // total loss
    out[base + 1] = commit;    // commitment
    out[base + 2] = entropy;   // entropy
}

extern "C" void kernel_launch(void* const* d_in, const int* in_sizes, int n_in,
                              void* d_out, int out_size, void* d_ws, size_t ws_size,
                              hipStream_t stream) {
    const float* z  = (const float*)d_in[0];   // (8,1024,1024) f32
    const float* cb = (const float*)d_in[1];   // (8,4096,128)  f32
    float* out = (float*)d_out;
    float* ws  = (float*)d_ws;

    mvq_zero_ws<<<(2 + NHEAD * KCODE + 255) / 256, 256, 0, stream>>>(ws);
    mvq_e2<<<(NHEAD * KCODE) / 8, 256, 0, stream>>>(cb, ws);
    dim3 grid(T_TOT / 128, NHEAD);
    mvq_main<<<grid, 256, 0, stream>>>(z, cb, out, ws);
    mvq_entropy<<<NHEAD, 256, 0, stream>>>(ws);
    mvq_finalize<<<1, 1, 0, stream>>>(ws, out);
}